// PointPillarsScatter_15006615733710
// MI455X (gfx1250) — compile-verified
//
#include <hip/hip_runtime.h>

// PointPillars scatter, MI455X (gfx1250, wave32).
// Strategy: (1) winner-index map via atomicMax (last-write-wins == max pillar
// index wins), 4 MB, L2-resident; (2) dense, fully-coalesced gather pass that
// writes the [B,C,H,W] output with float4 stores, using float4 row loads +
// 4x4 in-register transpose to minimize vmem instruction count.

namespace {
constexpr int B = 4;
constexpr int C = 64;
constexpr int H = 512;
constexpr int W = 512;
constexpr int P = 409600;
constexpr int NCELL = B * H * W;      // 1,048,576 BEV cells (winner map entries)
constexpr int WQ = W / 4;             // 128 float4 groups per row
}

// ---------------------------------------------------------------------------
// Kernel 1: init winner map to -1 (int4 = b128 stores, 1024 blocks x 256).
// ---------------------------------------------------------------------------
__global__ void pp_init_winner(int4* __restrict__ win4) {
  int t = blockIdx.x * blockDim.x + threadIdx.x;    // 0 .. NCELL/4-1
  win4[t] = make_int4(-1, -1, -1, -1);
}

// ---------------------------------------------------------------------------
// Kernel 2: per-pillar atomicMax of pillar index into its BEV cell.
// coords row = [b, z, y, x] int32 -> one aligned int4 load per pillar.
// y/x range-checked; out-of-range b also dropped (reference drops anything
// routing outside [0, B*H*W)).
// ---------------------------------------------------------------------------
__global__ void pp_scatter_winner(const int4* __restrict__ coords,
                                  int* __restrict__ win) {
  int p = blockIdx.x * blockDim.x + threadIdx.x;
  if (p >= P) return;
  int4 cd = coords[p];                 // {b, z, y, x}
  int b = cd.x, y = cd.z, x = cd.w;
  if ((unsigned)y < (unsigned)H && (unsigned)x < (unsigned)W &&
      (unsigned)b < (unsigned)B) {
    int flat = (b * H + y) * W + x;    // 0 .. NCELL-1
    atomicMax(&win[flat], p);          // global_atomic_max_i32, no return
  }
}

// ---------------------------------------------------------------------------
// Kernel 3: dense gather. One thread owns 4 consecutive cells (same b,y;
// x = 4*x4 .. 4*x4+3). Loads the int4 winner group (coalesced b128),
// prefetches the four 256 B feature rows, then walks channels 4 at a time:
// one float4 load per row (b128, 16 B/lane) + 4x4 register transpose +
// 4 coalesced float4 stores into [B,C,H,W]. Empty cells (p < 0) produce
// 0.0f via EXEC-masked loads + select.
// ---------------------------------------------------------------------------
__global__ void pp_gather_out(const float* __restrict__ vox,
                              const int4* __restrict__ win4,
                              float4* __restrict__ out) {
  int t = blockIdx.x * blockDim.x + threadIdx.x;    // 0 .. NCELL/4-1
  int x4 = t & (WQ - 1);                            // bits [6:0]
  int r  = t >> 7;
  int y  = r & (H - 1);                             // bits [8:0] of r
  int b  = r >> 9;

  int4 p4 = win4[t];                                // winners for 4 cells
  bool v0 = p4.x >= 0, v1 = p4.y >= 0, v2 = p4.z >= 0, v3 = p4.w >= 0;

  const float* r0 = vox + (size_t)p4.x * C;         // 256 B row, 16 B aligned
  const float* r1 = vox + (size_t)p4.y * C;
  const float* r2 = vox + (size_t)p4.z * C;
  const float* r3 = vox + (size_t)p4.w * C;

  // Pull each winning 256 B feature row toward the WGP before the channel
  // loop consumes it (global_prefetch_b8; two per row in case of 128 B
  // lines). 100 MB of features fit in the 192 MB L2, so re-reads are cheap.
  if (v0) { __builtin_prefetch(r0, 0, 3); __builtin_prefetch(r0 + 32, 0, 3); }
  if (v1) { __builtin_prefetch(r1, 0, 3); __builtin_prefetch(r1 + 32, 0, 3); }
  if (v2) { __builtin_prefetch(r2, 0, 3); __builtin_prefetch(r2 + 32, 0, 3); }
  if (v3) { __builtin_prefetch(r3, 0, 3); __builtin_prefetch(r3 + 32, 0, 3); }

  // Output float4 index for (b, c, y, x4): ((b*C + c)*H + y)*WQ + x4.
  int obase = ((b * C) * H + y) * WQ + x4;          // c = 0 slot
  const int ostride = H * WQ;                       // 65,536 float4 per channel

  const float4 z4 = make_float4(0.0f, 0.0f, 0.0f, 0.0f);

#pragma unroll 4
  for (int c = 0; c < C; c += 4) {
    // One b128 load per feature row covering channels c..c+3.
    float4 f0 = v0 ? *(const float4*)(r0 + c) : z4;
    float4 f1 = v1 ? *(const float4*)(r1 + c) : z4;
    float4 f2 = v2 ? *(const float4*)(r2 + c) : z4;
    float4 f3 = v3 ? *(const float4*)(r3 + c) : z4;

    // 4x4 register transpose: channel k's store = that channel's value for
    // the 4 consecutive x cells (rows r0..r3).
    out[obase + (c + 0) * ostride] = make_float4(f0.x, f1.x, f2.x, f3.x);
    out[obase + (c + 1) * ostride] = make_float4(f0.y, f1.y, f2.y, f3.y);
    out[obase + (c + 2) * ostride] = make_float4(f0.z, f1.z, f2.z, f3.z);
    out[obase + (c + 3) * ostride] = make_float4(f0.w, f1.w, f2.w, f3.w);
  }
}

// ---------------------------------------------------------------------------
// Launch wrapper. d_in[0] = voxel_features [P, C] fp32,
//                 d_in[1] = coords [P, 4] int32,
//                 d_out   = [B, C, H, W] fp32,
//                 d_ws    = winner map, NCELL * 4 = 4 MB.
// ---------------------------------------------------------------------------
extern "C" void kernel_launch(void* const* d_in, const int* in_sizes, int n_in,
                              void* d_out, int out_size, void* d_ws, size_t ws_size,
                              hipStream_t stream) {
  (void)in_sizes; (void)n_in; (void)out_size; (void)ws_size;

  const float* vox    = (const float*)d_in[0];
  const int4*  coords = (const int4*)d_in[1];
  int*         win    = (int*)d_ws;          // NCELL int32 (4 MB)
  float4*      out    = (float4*)d_out;

  const int threads = 256;                   // 8 wave32 per block

  // 1) winner map <- -1 (NCELL/4 int4 stores)
  pp_init_winner<<<(NCELL / 4) / threads, threads, 0, stream>>>((int4*)win);

  // 2) atomicMax pillar index per cell
  pp_scatter_winner<<<(P + threads - 1) / threads, threads, 0, stream>>>(coords, win);

  // 3) dense coalesced gather into [B, C, H, W]
  pp_gather_out<<<(NCELL / 4) / threads, threads, 0, stream>>>(vox, (const int4*)win, out);
}